// FidelityKernel_627065225216
// MI455X (gfx1250) — compile-verified
//
#include <hip/hip_runtime.h>
#include <hip/hip_bf16.h>

typedef __attribute__((ext_vector_type(2))) float v2f;
typedef __attribute__((ext_vector_type(8))) float v8f;

#define M_MODES   12
#define N_PHOT    6
#define PADC      8          // photon index padded 6 -> 8
#define NSAMP     384
#define NCOLS     (NSAMP * PADC)   // 3072 rows/cols of the big complex GEMM
#define PAIR_STRIDE 74       // floats per pair in LDS (36 cplx = 72, +2 pad, odd float2 stride)

// ---------------------------------------------------------------------------
// Kernel 1: build L = conj(U2)^T slices and R = U1 slices, layout [NCOLS][12].
//   R[(i*8+b)][k] = U1[i][k][b]        = sum_m A[k][m] e^{+i x1[i][m]} B[m][b]
//   L[(j*8+a)][k] = conj(U2[j][k][a])  = sum_m conj(A[k][m]) e^{-i x2[j][m]} conj(B[m][a])
// Padding entries (b>=6 / a>=6) are written as zeros.
// ---------------------------------------------------------------------------
__global__ __launch_bounds__(96)
void fk_prep(const float* __restrict__ x1, const float* __restrict__ x2,
             const float* __restrict__ Are, const float* __restrict__ Aim,
             const float* __restrict__ Bre, const float* __restrict__ Bim,
             float* __restrict__ Lre, float* __restrict__ Lim,
             float* __restrict__ Rre, float* __restrict__ Rim)
{
    __shared__ float p1r[M_MODES], p1i[M_MODES], p2r[M_MODES], p2i[M_MODES];
    const int s = blockIdx.x;
    const int t = threadIdx.x;

    if (t < M_MODES) {
        float sn, cs;
        __sincosf(x1[s * M_MODES + t], &sn, &cs);
        p1r[t] = cs; p1i[t] = sn;                 // e^{+i x1}
    } else if (t < 2 * M_MODES) {
        const int m = t - M_MODES;
        float sn, cs;
        __sincosf(x2[s * M_MODES + m], &sn, &cs);
        p2r[m] = cs; p2i[m] = -sn;                // e^{-i x2} (conjugated phase)
    }
    __syncthreads();

    const int k = t >> 3;        // 0..11
    const int c = t & 7;         // 0..7 (padded photon index)

    float er = 0.f, ei = 0.f, fr = 0.f, fi = 0.f;
    if (c < N_PHOT) {
#pragma unroll
        for (int m = 0; m < M_MODES; ++m) {
            const float ar = Are[k * M_MODES + m], ai = Aim[k * M_MODES + m];
            const float br = Bre[m * M_MODES + c], bi = Bim[m * M_MODES + c];
            // R side: A[k][m] * ph1[m] * B[m][c]
            const float t1r = ar * p1r[m] - ai * p1i[m];
            const float t1i = ar * p1i[m] + ai * p1r[m];
            er = fmaf(t1r, br, fmaf(-t1i, bi, er));
            ei = fmaf(t1r, bi, fmaf( t1i, br, ei));
            // L side: conj(A[k][m]) * ph2c[m] * conj(B[m][c])
            const float t2r = fmaf(ar, p2r[m],  ai * p2i[m]);
            const float t2i = fmaf(ar, p2i[m], -ai * p2r[m]);
            fr = fmaf(t2r, br, fmaf( t2i, bi, fr));
            fi = fmaf(t2i, br, fmaf(-t2r, bi, fi));
        }
    }
    const int idx = (s * PADC + c) * M_MODES + k;
    Rre[idx] = er; Rim[idx] = ei;
    Lre[idx] = fr; Lim[idx] = fi;
}

// ---------------------------------------------------------------------------
// Kernel 2: per block: 16 i x 8 j = 128 pairs.
// Phase A: complex GEMM W = L * R via V_WMMA_F32_16X16X4_F32 (4 real chains),
//          one 16-row tile strip per wave, 8 tile-cols, K=12 in 3 steps.
// Phase B: thread t computes permanent of pair t (6x6 complex, Glynn/Gray).
// ---------------------------------------------------------------------------
__global__ __launch_bounds__(128)
void fk_pairperm(const float* __restrict__ Lre, const float* __restrict__ Lim,
                 const float* __restrict__ Rre, const float* __restrict__ Rim,
                 float* __restrict__ out)
{
    __shared__ float lds[128 * PAIR_STRIDE];

    const int iblk = blockIdx.x;           // 0..23  (16 i's each)
    const int jblk = blockIdx.y;           // 0..47  ( 8 j's each)
    const int t    = threadIdx.x;          // 0..127
    const int wave = t >> 5;               // tile-row strip (0..3)
    const int lane = t & 31;
    const int l16  = lane & 15;
    const int hi   = lane >> 4;            // lane half
    const int ko   = hi * 2;               // per-lane K offset inside a 4-step

    // ---- Phase A: WMMA complex GEMM ----
    // A fragments (j side), fixed for this wave: row = jblk*64 + wave*16 + M
    const int lrow = jblk * 64 + wave * 16 + l16;
    const float* lreP = Lre + lrow * M_MODES;
    const float* limP = Lim + lrow * M_MODES;
    v2f aRe[3], aIm[3];
#pragma unroll
    for (int s = 0; s < 3; ++s) {
        const int k = s * 4 + ko;
        aRe[s] = v2f{lreP[k], lreP[k + 1]};
        aIm[s] = v2f{limP[k], limP[k + 1]};
    }

    for (int tc = 0; tc < 8; ++tc) {
        const int col = iblk * 128 + tc * 16 + l16;
        const float* rreP = Rre + col * M_MODES;
        const float* rimP = Rim + col * M_MODES;

        v8f acRR = {}, acII = {}, acRI = {}, acIR = {};
#pragma unroll
        for (int s = 0; s < 3; ++s) {
            const int k = s * 4 + ko;
            const v2f bRe = v2f{rreP[k], rreP[k + 1]};
            const v2f bIm = v2f{rimP[k], rimP[k + 1]};
            acRR = __builtin_amdgcn_wmma_f32_16x16x4_f32(false, aRe[s], false, bRe, (short)0, acRR, false, false);
            acII = __builtin_amdgcn_wmma_f32_16x16x4_f32(false, aIm[s], false, bIm, (short)0, acII, false, false);
            acRI = __builtin_amdgcn_wmma_f32_16x16x4_f32(false, aRe[s], false, bIm, (short)0, acRI, false, false);
            acIR = __builtin_amdgcn_wmma_f32_16x16x4_f32(false, aIm[s], false, bRe, (short)0, acIR, false, false);
        }

        // Scatter valid entries to LDS. Lane holds column n=l16 of the tile,
        // rows M = r + 8*hi (C/D layout).
        const int cloc    = tc * 16 + l16;     // col within block (0..127)
        const int b       = cloc & 7;
        const int i_local = cloc >> 3;         // 0..15
        if (b < N_PHOT) {
#pragma unroll
            for (int r = 0; r < 8; ++r) {
                const int rloc = wave * 16 + r + 8 * hi;   // row within block (0..63)
                const int a = rloc & 7;
                if (a < N_PHOT) {
                    const int j_local = rloc >> 3;         // 0..7
                    const int pair = i_local * 8 + j_local;
                    const int off  = pair * PAIR_STRIDE + (a * N_PHOT + b) * 2;
                    lds[off]     = acRR[r] - acII[r];       // Re(W)
                    lds[off + 1] = acRI[r] + acIR[r];       // Im(W)
                }
            }
        }
    }
    __syncthreads();

    // ---- Phase B: Glynn permanent (Gray code over 32 sign vectors) ----
    {
        const int i_local = t >> 3;
        const int j_local = t & 7;
        const float* W = &lds[t * PAIR_STRIDE];

        float wr[N_PHOT][N_PHOT], wi[N_PHOT][N_PHOT];
#pragma unroll
        for (int a = 0; a < N_PHOT; ++a)
#pragma unroll
            for (int b = 0; b < N_PHOT; ++b) {
                wr[a][b] = W[(a * N_PHOT + b) * 2];
                wi[a][b] = W[(a * N_PHOT + b) * 2 + 1];
            }

        // rowsums for all-plus sign vector
        float rr[N_PHOT], ri[N_PHOT];
#pragma unroll
        for (int b = 0; b < N_PHOT; ++b) {
            float sr = 0.f, si = 0.f;
#pragma unroll
            for (int a = 0; a < N_PHOT; ++a) { sr += wr[a][b]; si += wi[a][b]; }
            rr[b] = sr; ri[b] = si;
        }

        // product over columns
        float pr = rr[0], pi = ri[0];
#pragma unroll
        for (int b = 1; b < N_PHOT; ++b) {
            const float nr = pr * rr[b] - pi * ri[b];
            const float ni = pr * ri[b] + pi * rr[b];
            pr = nr; pi = ni;
        }
        float accR = pr, accI = pi;

        unsigned g = 0;
        float par = 1.f;
        for (int k2 = 1; k2 < 32; ++k2) {
            const int p = __builtin_ctz((unsigned)k2);  // flipped Gray bit -> row p+1
            g ^= (1u << p);
            const float sgn = ((g >> p) & 1u) ? -2.f : 2.f;
            const int a = p + 1;
#pragma unroll
            for (int b = 0; b < N_PHOT; ++b) {
                rr[b] = fmaf(sgn, wr[a][b], rr[b]);
                ri[b] = fmaf(sgn, wi[a][b], ri[b]);
            }
            par = -par;
            pr = rr[0]; pi = ri[0];
#pragma unroll
            for (int b = 1; b < N_PHOT; ++b) {
                const float nr = pr * rr[b] - pi * ri[b];
                const float ni = pr * ri[b] + pi * rr[b];
                pr = nr; pi = ni;
            }
            accR = fmaf(par, pr, accR);
            accI = fmaf(par, pi, accI);
        }

        // perm = acc/32 ; K = |perm|^2
        const float K = (accR * accR + accI * accI) * (1.0f / 1024.0f);
        const int gi = iblk * 16 + i_local;
        const int gj = jblk * 8 + j_local;
        out[gi * NSAMP + gj] = K;
    }
}

// ---------------------------------------------------------------------------
extern "C" void kernel_launch(void* const* d_in, const int* in_sizes, int n_in,
                              void* d_out, int out_size, void* d_ws, size_t ws_size,
                              hipStream_t stream)
{
    (void)in_sizes; (void)n_in; (void)out_size; (void)ws_size;
    const float* x1  = (const float*)d_in[0];
    const float* x2  = (const float*)d_in[1];
    const float* Are = (const float*)d_in[2];
    const float* Aim = (const float*)d_in[3];
    const float* Bre = (const float*)d_in[4];
    const float* Bim = (const float*)d_in[5];

    float* ws = (float*)d_ws;
    const size_t SEG = (size_t)NCOLS * M_MODES;   // 36864 floats
    float* Lre = ws + 0 * SEG;
    float* Lim = ws + 1 * SEG;
    float* Rre = ws + 2 * SEG;
    float* Rim = ws + 3 * SEG;

    fk_prep<<<NSAMP, 96, 0, stream>>>(x1, x2, Are, Aim, Bre, Bim, Lre, Lim, Rre, Rim);

    dim3 grid(NSAMP / 16, NSAMP / 8);             // 24 x 48
    fk_pairperm<<<grid, 128, 0, stream>>>(Lre, Lim, Rre, Rim, (float*)d_out);
}